// GRNN_90013924590090
// MI455X (gfx1250) — compile-verified
//
#include <hip/hip_runtime.h>
#include <hip/hip_bf16.h>

// GRNN: 2-iteration gated graph GRU, 500k nodes x DIM=10, 8M edges.
// Memory/atomic-bound (~1.5 GB traffic; whole working set fits in 192MB L2).
// FP32 throughout to match reference; matmuls via CDNA5 V_WMMA_F32_16X16X4_F32
// (wave32, one 16-node tile per wave, K padded 10->12, N padded 10->16).
// Weights staged zero-padded in LDS so B operands are branch-free ds_load_b64.

namespace {
constexpr int kNodes = 500000;
constexpr int kEdges = 8000000;
constexpr int kDim   = 10;
constexpr int kIters = 2;
constexpr int kTiles = kNodes / 16;  // 31250, exact
}

typedef __attribute__((ext_vector_type(2))) float v2f;
typedef __attribute__((ext_vector_type(8))) float v8f;

// ---------------------------------------------------------------- zero x ----
__global__ __launch_bounds__(256) void grnn_zero_kernel(float4* __restrict__ x, int n4) {
  int i = blockIdx.x * blockDim.x + threadIdx.x;
  if (i < n4) x[i] = make_float4(0.f, 0.f, 0.f, 0.f);
}

// ---------------------------------------------------- edge gather+scatter ---
// x[recv] += h[send] for edges whose sender is active at this iteration
// (equivalent to the reference's h = where(active, h, 0) before segment_sum).
__global__ __launch_bounds__(256) void grnn_edge_kernel(
    const int* __restrict__ edges, const int* __restrict__ depth,
    const float* __restrict__ h, float* __restrict__ x, int iter) {
  int e = blockIdx.x * blockDim.x + threadIdx.x;
  if (e >= kEdges) return;
  int rv = edges[e];           // edges[0][e]
  int sd = edges[kEdges + e];  // edges[1][e]
  if (depth[sd] + iter > kIters) return;  // inactive sender contributes 0
  // rows are 40B => always 8B aligned: 5x global_load_b64
  const float2* hs = reinterpret_cast<const float2*>(h) + (size_t)sd * 5;
  float* xr = x + (size_t)rv * kDim;
#pragma unroll
  for (int k = 0; k < 5; ++k) {
    float2 v = hs[k];
    __hip_atomic_fetch_add(xr + 2 * k,     v.x, __ATOMIC_RELAXED, __HIP_MEMORY_SCOPE_AGENT);
    __hip_atomic_fetch_add(xr + 2 * k + 1, v.y, __ATOMIC_RELAXED, __HIP_MEMORY_SCOPE_AGENT);
  }
}

// ------------------------------------------------------------- node GRU -----
// acc += A(16x12) * W^T(12x16) as 3 chained f32 WMMAs (K=4 each).
// B layout (4x16, 2 VGPRs): N = lane&15, K = vgprIdx + 2*(lane>=16).
// Weight tile is LDS-resident, padded [n*16+k] => operand = one ds_load_b64.
__device__ __forceinline__ v8f grnn_mm3(v8f acc, const v2f a[3],
                                        const float* wp, int n, int hh) {
#pragma unroll
  for (int c = 0; c < 3; ++c) {
    int k0 = 4 * c + 2 * hh;  // even => 8B-aligned LDS address
    float2 bv = *reinterpret_cast<const float2*>(wp + n * 16 + k0);
    v2f b;
    b.x = bv.x;
    b.y = bv.y;
    acc = __builtin_amdgcn_wmma_f32_16x16x4_f32(false, a[c], false, b,
                                                (short)0, acc, false, false);
  }
  return acc;
}

__device__ __forceinline__ float grnn_sigmoid(float p) {
  return 1.0f / (1.0f + __expf(-p));
}

__global__ __launch_bounds__(256) void grnn_node_kernel(
    const float* __restrict__ x, const float* __restrict__ h,
    const int* __restrict__ depth,
    const float* __restrict__ Wz, const float* __restrict__ bz,
    const float* __restrict__ Uz, const float* __restrict__ buz,
    const float* __restrict__ Wr, const float* __restrict__ br,
    const float* __restrict__ Ur, const float* __restrict__ bur,
    const float* __restrict__ Wh, const float* __restrict__ bh,
    const float* __restrict__ Uh, const float* __restrict__ buh,
    float* __restrict__ hout, int iter) {
  __shared__ float wpad[6 * 256];  // {Wz,Uz,Wr,Ur,Wh,Uh} padded 16x16, [n*16+k]
  __shared__ float bpad[3 * 16];   // fused biases {bz+buz, br+bur, bh+buh}
  __shared__ float rhT[8 * 256];   // per-wave 16x16 r*h staging (1KB each)

  // ---- stage weights & biases (all 256 threads, branch-light) ----
  {
    const int t  = threadIdx.x;
    const int n0 = t >> 4, ks = t & 15;
    const bool ok = (n0 < kDim) && (ks < kDim);
    const int  src = (n0 < kDim ? n0 : 0) * kDim + (ks < kDim ? ks : 0);
    const float* Ws[6] = {Wz, Uz, Wr, Ur, Wh, Uh};
#pragma unroll
    for (int w = 0; w < 6; ++w) wpad[w * 256 + t] = ok ? Ws[w][src] : 0.f;
    if (t < 16) {
      const int tc = (t < kDim) ? t : 0;
      const float bm = (t < kDim) ? 1.f : 0.f;
      bpad[t]      = (bz[tc] + buz[tc]) * bm;
      bpad[16 + t] = (br[tc] + bur[tc]) * bm;
      bpad[32 + t] = (bh[tc] + buh[tc]) * bm;
    }
  }
  __syncthreads();

  const int lane = threadIdx.x & 31;
  const int wave = threadIdx.x >> 5;
  const int tile = blockIdx.x * 8 + wave;
  if (tile >= kTiles) return;          // wave-uniform: EXEC all-1s at WMMAs
  const long base = (long)tile * 16;   // all tiles full (500000 = 16*31250)
  const int hh = lane >> 4;            // lane half
  const int n  = lane & 15;            // N / A-row index
  float* myLds = rhT + wave * 256;

  // ---- A-layout loads: row = lane&15, K = vgpr + 2*half (branch-free) ----
  const long ga = base + n;
  const float amask = (depth[ga] + iter <= kIters) ? 1.f : 0.f;
  v2f xa[3], ha[3];
#pragma unroll
  for (int c = 0; c < 3; ++c) {
    const int   k0  = 4 * c + 2 * hh;           // 0,2,4,6,8,10 (even => 8B ok)
    const int   k0c = (k0 < kDim) ? k0 : 8;     // clamp pad chunk in-bounds
    const float km  = (k0 < kDim) ? 1.f : 0.f;  // zero the K=10,11 pad
    float2 xv = *reinterpret_cast<const float2*>(x + ga * kDim + k0c);
    float2 hv = *reinterpret_cast<const float2*>(h + ga * kDim + k0c);
    xa[c].x = xv.x * km;           xa[c].y = xv.y * km;
    ha[c].x = hv.x * (km * amask); ha[c].y = hv.y * (km * amask);
  }

  // ---- C/D-layout h (masked): M = v + 8*half, N = lane&15 (branch-free) ----
  const int   ncol  = (n < kDim) ? n : kDim - 1;
  const float nmask = (n < kDim) ? 1.f : 0.f;
  v8f hcd;
  float cmask[8];
#pragma unroll
  for (int v = 0; v < 8; ++v) {
    const long m  = base + v + 8 * hh;
    const float mk = (depth[m] + iter <= kIters) ? 1.f : 0.f;
    hcd[v]   = h[m * kDim + ncol] * (mk * nmask);
    cmask[v] = mk;
  }

  // ---- z gate ----
  v8f accz;
  {
    const float bv = bpad[n];
#pragma unroll
    for (int v = 0; v < 8; ++v) accz[v] = bv;
  }
  accz = grnn_mm3(accz, xa, wpad + 0 * 256, n, hh);
  accz = grnn_mm3(accz, ha, wpad + 1 * 256, n, hh);
  float zg[8];
#pragma unroll
  for (int v = 0; v < 8; ++v) zg[v] = grnn_sigmoid(accz[v]);

  // ---- r gate ----
  v8f accr;
  {
    const float bv = bpad[16 + n];
#pragma unroll
    for (int v = 0; v < 8; ++v) accr[v] = bv;
  }
  accr = grnn_mm3(accr, xa, wpad + 2 * 256, n, hh);
  accr = grnn_mm3(accr, ha, wpad + 3 * 256, n, hh);

  // ---- r*h: C/D layout -> LDS -> A layout (same-wave DS ops in-order) ----
#pragma unroll
  for (int v = 0; v < 8; ++v)
    myLds[(v + 8 * hh) * 16 + n] = grnn_sigmoid(accr[v]) * hcd[v];
  v2f rha[3];
#pragma unroll
  for (int c = 0; c < 3; ++c) {
    const int k0 = 4 * c + 2 * hh;  // <= 10; cols 10..15 hold exact zeros
    float2 rv = *reinterpret_cast<const float2*>(myLds + n * 16 + k0);
    rha[c].x = rv.x;
    rha[c].y = rv.y;
  }

  // ---- candidate state ----
  v8f acch;
  {
    const float bv = bpad[32 + n];
#pragma unroll
    for (int v = 0; v < 8; ++v) acch[v] = bv;
  }
  acch = grnn_mm3(acch, xa,  wpad + 4 * 256, n, hh);
  acch = grnn_mm3(acch, rha, wpad + 5 * 256, n, hh);

  // ---- h_new = z*h + (1-z)*tanh(pre), zeroed for inactive nodes ----
  if (n < kDim) {  // single divergent region
#pragma unroll
    for (int v = 0; v < 8; ++v) {
      const long m = base + v + 8 * hh;
      const float t  = tanhf(acch[v]);
      const float hn = zg[v] * hcd[v] + (1.f - zg[v]) * t;
      hout[m * kDim + n] = hn * cmask[v];
    }
  }
}

// ---------------------------------------------------------------------------
extern "C" void kernel_launch(void* const* d_in, const int* in_sizes, int n_in,
                              void* d_out, int out_size, void* d_ws, size_t ws_size,
                              hipStream_t stream) {
  const float* h_in  = (const float*)d_in[0];
  const int*   edges = (const int*)d_in[1];
  const int*   depth = (const int*)d_in[2];
  const float* Wz  = (const float*)d_in[3];
  const float* bz  = (const float*)d_in[4];
  const float* Uz  = (const float*)d_in[5];
  const float* buz = (const float*)d_in[6];
  const float* Wr  = (const float*)d_in[7];
  const float* br  = (const float*)d_in[8];
  const float* Ur  = (const float*)d_in[9];
  const float* bur = (const float*)d_in[10];
  const float* Wh  = (const float*)d_in[11];
  const float* bh  = (const float*)d_in[12];
  const float* Uh  = (const float*)d_in[13];
  const float* buh = (const float*)d_in[14];

  float* x    = (float*)d_ws;   // 500k x 10 f32 segment-sum accumulator (20MB)
  float* hout = (float*)d_out;

  const int n4 = kNodes * kDim / 4;  // 1,250,000 (exact)
  for (int iter = 0; iter < kIters; ++iter) {
    const float* hcur = (iter == 0) ? h_in : hout;  // iter1 updates d_out row-locally
    grnn_zero_kernel<<<(n4 + 255) / 256, 256, 0, stream>>>((float4*)x, n4);
    grnn_edge_kernel<<<(kEdges + 255) / 256, 256, 0, stream>>>(edges, depth, hcur, x, iter);
    grnn_node_kernel<<<(kTiles + 7) / 8, 256, 0, stream>>>(
        x, hcur, depth, Wz, bz, Uz, buz, Wr, br, Ur, bur, Wh, bh, Uh, buh, hout, iter);
  }
}